// BahdanauAttention_6176162972448
// MI455X (gfx1250) — compile-verified
//
#include <hip/hip_runtime.h>
#include <math.h>

#define B_ 8
#define T_ 512
#define S_ 512
#define H_ 128

typedef __attribute__((ext_vector_type(2))) float v2f;
typedef __attribute__((ext_vector_type(8))) float v8f;

// gfx1250 hardware transcendental tanh (VOP1 v_tanh_f32, TRANS pipe,
// co-executes with VALU). 268M of these dominate this workload.
__device__ __forceinline__ float fast_tanh(float x) {
  float r;
  asm volatile("v_tanh_f32 %0, %1" : "=v"(r) : "v"(x));
  return r;
}

// CDNA5 async copy: global -> LDS, 16B per lane, tracked by ASYNCcnt.
// lds_off is the LDS byte offset (low 32 bits of a generic LDS pointer).
__device__ __forceinline__ void async_g2l_b128(unsigned lds_off, const float* gptr) {
  asm volatile("global_load_async_to_lds_b128 %0, %1, off"
               :: "v"(lds_off), "v"(gptr)
               : "memory");
}

// ---------------------------------------------------------------------------
// WMMA f32 16x16x4 tile GEMM helpers.
// A 16x4 layout: lane l holds A[m=l&15][k = 2*(l>>4) + j], j = VGPR idx (contig float2)
// B 4x16 layout: lane l holds B[k = 2*(l>>4) + r][n=l&15],  r = VGPR idx
// C/D:           lane l, VGPR r -> D[m = r + 8*(l>>4)][n = l&15]
// ---------------------------------------------------------------------------

// Y(M x 128) = X(M x 128) @ W(128 x 128)^T   (row-major everywhere)
// grid.x = M/16 row tiles, 256 threads = 8 waves, wave w owns col tile w*16.
__global__ __launch_bounds__(256) void proj_wmma_kernel(
    const float* __restrict__ X, const float* __restrict__ W,
    float* __restrict__ Y) {
  constexpr int K = 128, N = 128, KC = 64, APAD = 68;
  __shared__ float Alds[16 * APAD];    //  ~4.3 KB
  __shared__ float Wlds[128 * APAD];   // ~34.8 KB
  const int tid = threadIdx.x;
  const int wid = tid >> 5, lane = tid & 31;
  const int ln = lane & 15, half = lane >> 4;
  const int mb = blockIdx.x * 16;
  const int nb = wid * 16;
  v8f acc = {};
  for (int kc = 0; kc < K; kc += KC) {
    __syncthreads();
#pragma unroll
    for (int i = 0; i < 4; ++i) {          // A tile 16x64, coalesced
      int idx = tid + i * 256;
      int r = idx >> 6, c = idx & 63;
      Alds[r * APAD + c] = X[(mb + r) * K + kc + c];
    }
#pragma unroll
    for (int i = 0; i < 32; ++i) {         // W chunk 128x64, coalesced
      int idx = tid + i * 256;
      int r = idx >> 6, c = idx & 63;
      Wlds[r * APAD + c] = W[r * K + kc + c];
    }
    __syncthreads();
#pragma unroll
    for (int kk = 0; kk < KC; kk += 4) {
      v2f a = *(const v2f*)&Alds[ln * APAD + kk + 2 * half];
      v2f b = *(const v2f*)&Wlds[(nb + ln) * APAD + kk + 2 * half];
      acc = __builtin_amdgcn_wmma_f32_16x16x4_f32(false, a, false, b,
                                                  (short)0, acc, false, false);
    }
  }
#pragma unroll
  for (int r = 0; r < 8; ++r)
    Y[(size_t)(mb + r + 8 * half) * N + nb + ln] = acc[r];
}

// scores[b, t, s] = sum_h v[h] * tanh(qt[b,t,h] + et[b,s,h])
// grid = (T/16, B). qt tile + v staged once; et streamed in 32-s chunks via
// CDNA5 async global->LDS copies, double-buffered so the copy of chunk i+1
// overlaps the 256-tanh-per-thread compute of chunk i. ASYNCcnt completes
// in order, so s_wait_asynccnt<=4 retires exactly the previous chunk.
__global__ __launch_bounds__(256) void scores_kernel(
    const float* __restrict__ qt, const float* __restrict__ et,
    const float* __restrict__ v, float* __restrict__ scores_out) {
  constexpr int QPAD = 132, CH = 32, NCHUNK = S_ / CH;
  __shared__ float qlds[16 * QPAD];       //  ~8.4 KB
  __shared__ float elds[2][CH * QPAD];    // ~33.8 KB double buffer
  __shared__ float vlds[H_];
  const int tid = threadIdx.x;
  const int b = blockIdx.y;
  const int t0 = blockIdx.x * 16;

  // issue async prefetch of chunk 0 into buffer 0 (4 b128 per thread)
  {
    const float* src = et + (size_t)(b * S_) * H_;
#pragma unroll
    for (int i = 0; i < 4; ++i) {
      int f4 = tid + i * 256;             // 1024 float4 per chunk
      int r = f4 >> 5, c4 = (f4 & 31) * 4;
      async_g2l_b128((unsigned)(uintptr_t)&elds[0][r * QPAD + c4],
                     src + (size_t)r * H_ + c4);
    }
  }
#pragma unroll
  for (int i = 0; i < 8; ++i) {           // qt tile 16x128, coalesced
    int idx = tid + i * 256;
    int r = idx >> 7, c = idx & 127;
    qlds[r * QPAD + c] = qt[(size_t)(b * T_ + t0 + r) * H_ + c];
  }
  if (tid < H_) vlds[tid] = v[tid];

  const int t = tid & 15;
  const int sl = tid >> 4;                // 0..15 -> s = sl, sl+16
  for (int ch = 0; ch < NCHUNK; ++ch) {
    const int cur = ch & 1;
    if (ch + 1 < NCHUNK) {
      // prefetch next chunk into the other buffer
      const float* src = et + (size_t)(b * S_ + (ch + 1) * CH) * H_;
#pragma unroll
      for (int i = 0; i < 4; ++i) {
        int f4 = tid + i * 256;
        int r = f4 >> 5, c4 = (f4 & 31) * 4;
        async_g2l_b128((unsigned)(uintptr_t)&elds[cur ^ 1][r * QPAD + c4],
                       src + (size_t)r * H_ + c4);
      }
      asm volatile("s_wait_asynccnt 0x4" ::: "memory");  // chunk ch landed
    } else {
      asm volatile("s_wait_asynccnt 0x0" ::: "memory");
    }
    __syncthreads();                      // all waves' chunk data visible

    float acc0 = 0.f, acc1 = 0.f;
    const float* eb = elds[cur];
#pragma unroll 8
    for (int h = 0; h < H_; ++h) {        // adjacent-h loads merge to b128
      float q = qlds[t * QPAD + h];
      float vv = vlds[h];
      acc0 += vv * fast_tanh(q + eb[sl * QPAD + h]);
      acc1 += vv * fast_tanh(q + eb[(sl + 16) * QPAD + h]);
    }
    size_t row = (size_t)(b * T_ + t0 + t) * S_;
    scores_out[row + ch * CH + sl]      = acc0;
    scores_out[row + ch * CH + sl + 16] = acc1;
    __syncthreads();                      // guard buffer before ch+2 prefetch
  }
}

// masked softmax over s, in place. Wave per row; 16 elems per lane.
__global__ __launch_bounds__(256) void softmax_kernel(
    float* __restrict__ attw, const int* __restrict__ lens) {
  const int wid = threadIdx.x >> 5, lane = threadIdx.x & 31;
  const int row = blockIdx.x * 8 + wid;        // B*T rows
  const int len = lens[row / T_];
  float* p = attw + (size_t)row * S_;
  float vals[16];
  float m = -INFINITY;
#pragma unroll
  for (int i = 0; i < 16; ++i) {
    int s = lane + 32 * i;
    float val = (s < len) ? p[s] : -INFINITY;
    vals[i] = val;
    m = fmaxf(m, val);
  }
#pragma unroll
  for (int off = 16; off >= 1; off >>= 1) m = fmaxf(m, __shfl_xor(m, off, 32));
  float sum = 0.f;
#pragma unroll
  for (int i = 0; i < 16; ++i) {
    int s = lane + 32 * i;
    float e = (s < len) ? __expf(vals[i] - m) : 0.f;
    vals[i] = e;
    sum += e;
  }
#pragma unroll
  for (int off = 16; off >= 1; off >>= 1) sum += __shfl_xor(sum, off, 32);
  float inv = 1.f / sum;
#pragma unroll
  for (int i = 0; i < 16; ++i) p[lane + 32 * i] = vals[i] * inv;
}

// context[b] = attn[b] (T x S) @ enc[b] (S x H).  grid = (T/16, B).
__global__ __launch_bounds__(256) void context_wmma_kernel(
    const float* __restrict__ attw, const float* __restrict__ enc,
    float* __restrict__ ctx) {
  constexpr int KC = 64, APAD = 68, NPAD = 132;
  __shared__ float Alds[16 * APAD];   //  ~4.3 KB
  __shared__ float Blds[64 * NPAD];   // ~33.8 KB, k-major
  const int tid = threadIdx.x;
  const int wid = tid >> 5, lane = tid & 31;
  const int ln = lane & 15, half = lane >> 4;
  const int b = blockIdx.y;
  const int mb = blockIdx.x * 16;
  const int nb = wid * 16;
  v8f acc = {};
  for (int kc = 0; kc < S_; kc += KC) {
    __syncthreads();
#pragma unroll
    for (int i = 0; i < 4; ++i) {     // attn tile 16x64
      int idx = tid + i * 256;
      int r = idx >> 6, c = idx & 63;
      Alds[r * APAD + c] = attw[(size_t)(b * T_ + mb + r) * S_ + kc + c];
    }
#pragma unroll
    for (int i = 0; i < 32; ++i) {    // enc chunk 64(k) x 128(n)
      int idx = tid + i * 256;
      int r = idx >> 7, c = idx & 127;
      Blds[r * NPAD + c] = enc[(size_t)(b * S_ + kc + r) * H_ + c];
    }
    __syncthreads();
#pragma unroll
    for (int kk = 0; kk < KC; kk += 4) {
      v2f a = *(const v2f*)&Alds[ln * APAD + kk + 2 * half];
      v2f bb;
      bb.x = Blds[(kk + 2 * half) * NPAD + nb + ln];
      bb.y = Blds[(kk + 2 * half + 1) * NPAD + nb + ln];
      acc = __builtin_amdgcn_wmma_f32_16x16x4_f32(false, a, false, bb,
                                                  (short)0, acc, false, false);
    }
  }
#pragma unroll
  for (int r = 0; r < 8; ++r)
    ctx[(size_t)(b * T_ + mb + r + 8 * half) * H_ + nb + ln] = acc[r];
}

// attn_out = tanh([ctx | query] (4096 x 256) @ W_out(128 x 256)^T + b_out)
// Fused: phase 0 uses ctx with W_out[:, :128], phase 1 query with W_out[:, 128:],
// accumulated into the same WMMA accumulators. grid.x = 4096/16.
__global__ __launch_bounds__(256) void out_wmma_kernel(
    const float* __restrict__ ctx, const float* __restrict__ query,
    const float* __restrict__ Wout, const float* __restrict__ bout,
    float* __restrict__ out) {
  constexpr int KC = 64, APAD = 68;
  __shared__ float Alds[16 * APAD];
  __shared__ float Wlds[128 * APAD];
  const int tid = threadIdx.x;
  const int wid = tid >> 5, lane = tid & 31;
  const int ln = lane & 15, half = lane >> 4;
  const int mb = blockIdx.x * 16;
  const int nb = wid * 16;
  v8f acc = {};
  for (int pc = 0; pc < 4; ++pc) {
    const float* Xp = (pc < 2) ? ctx : query;
    const int wo = (pc < 2) ? 0 : 128;
    const int kc = (pc & 1) * KC;
    __syncthreads();
#pragma unroll
    for (int i = 0; i < 4; ++i) {
      int idx = tid + i * 256;
      int r = idx >> 6, c = idx & 63;
      Alds[r * APAD + c] = Xp[(size_t)(mb + r) * H_ + kc + c];
    }
#pragma unroll
    for (int i = 0; i < 32; ++i) {
      int idx = tid + i * 256;
      int r = idx >> 6, c = idx & 63;
      Wlds[r * APAD + c] = Wout[r * (2 * H_) + wo + kc + c];
    }
    __syncthreads();
#pragma unroll
    for (int kk = 0; kk < KC; kk += 4) {
      v2f a = *(const v2f*)&Alds[ln * APAD + kk + 2 * half];
      v2f b = *(const v2f*)&Wlds[(nb + ln) * APAD + kk + 2 * half];
      acc = __builtin_amdgcn_wmma_f32_16x16x4_f32(false, a, false, b,
                                                  (short)0, acc, false, false);
    }
  }
  const float bias = bout[nb + ln];
#pragma unroll
  for (int r = 0; r < 8; ++r)
    out[(size_t)(mb + r + 8 * half) * H_ + nb + ln] = fast_tanh(acc[r] + bias);
}

extern "C" void kernel_launch(void* const* d_in, const int* in_sizes, int n_in,
                              void* d_out, int out_size, void* d_ws, size_t ws_size,
                              hipStream_t stream) {
  const float* query = (const float*)d_in[0];   // (B,T,H)
  const float* enc   = (const float*)d_in[1];   // (B,S,H)
  const int*   lens  = (const int*)d_in[2];     // (B,)
  const float* Ws    = (const float*)d_in[3];   // (H,H)
  const float* Wh    = (const float*)d_in[4];   // (H,H)
  const float* v     = (const float*)d_in[5];   // (H,)
  const float* Wout  = (const float*)d_in[6];   // (H,2H)
  const float* bout  = (const float*)d_in[7];   // (H,)

  float* attn_out = (float*)d_out;                         // B*T*H
  float* attw     = (float*)d_out + (size_t)B_ * T_ * H_;  // B*T*S

  const size_t BTH = (size_t)B_ * T_ * H_;
  float* qt  = (float*)d_ws;           // 2 MB
  float* et  = qt + BTH;               // 2 MB
  float* ctx = et + BTH;               // 2 MB

  // 1) projections (WMMA f32)
  proj_wmma_kernel<<<dim3((B_ * T_) / 16), 256, 0, stream>>>(query, Ws, qt);
  proj_wmma_kernel<<<dim3((B_ * S_) / 16), 256, 0, stream>>>(enc, Wh, et);
  // 2) additive scores: hw tanh + async double-buffered et streaming
  scores_kernel<<<dim3(T_ / 16, B_), 256, 0, stream>>>(qt, et, v, attw);
  // 3) masked softmax in place
  softmax_kernel<<<dim3((B_ * T_) / 8), 256, 0, stream>>>(attw, lens);
  // 4) context GEMM (WMMA f32)
  context_wmma_kernel<<<dim3(T_ / 16, B_), 256, 0, stream>>>(attw, enc, ctx);
  // 5) fused concat-projection + bias + tanh (WMMA f32)
  out_wmma_kernel<<<dim3((B_ * T_) / 16), 256, 0, stream>>>(ctx, query, Wout, bout, attn_out);
}